// MMFRamanAmplifier_67018669687416
// MI455X (gfx1250) — compile-verified
//
#include <hip/hip_runtime.h>
#include <hip/hip_bf16.h>

typedef float v2f __attribute__((ext_vector_type(2)));
typedef float v8f __attribute__((ext_vector_type(8)));

#define NTHREADS 224   // 7 waves of 32
#define F_TOT    104   // pumps(4) + channels(100)
#define NSTATE   416   // F_TOT * MODES(4)

__global__ __launch_bounds__(NTHREADS)
void MMFRamanAmplifier_kernel(const float* __restrict__ x,       // (256,20)
                              const float* __restrict__ rr,      // (801,)
                              const float* __restrict__ oi,      // (4,4)
                              const float* __restrict__ sigwl,   // (100,)
                              const float* __restrict__ lc,      // (3,)
                              float* __restrict__ out)           // (256,100,4)
{
    __shared__ float gmat[112 * 104];   // zero-padded rows 104..111
    __shared__ float resp[801];
    __shared__ float Qp[104 * 16];      // B operand, cols 4..15 stay zero
    __shared__ float tf[104];
    __shared__ float losses[NSTATE];
    __shared__ float P0[NSTATE];
    __shared__ float acc[NSTATE];
    __shared__ float argv[NSTATE];
    __shared__ float oi_s[16];

    const int b    = blockIdx.x;
    const int tid  = threadIdx.x;
    const int lane = tid & 31;
    const int w    = tid >> 5;          // wave id 0..6
    const int l15  = lane & 15;
    const int hi   = (lane >= 16) ? 1 : 0;

    const float c0  = lc[0], c1 = lc[1], c2 = lc[2];
    const float LN  = 0.23025850929940458f;   // ln(10)/10
    const float DFv = 5.0e10f;                 // FS / N_RESP
    const float dz  = 50000.0f / 99.0f;

    // ---- load small tables / init state -------------------------------
    for (int e = tid; e < 801; e += NTHREADS) resp[e] = rr[e];
    if (tid < 16) oi_s[tid] = oi[tid];
    for (int e = tid; e < F_TOT; e += NTHREADS) {
        float wl = (e < 4) ? x[b * 20 + e] : sigwl[e - 4];
        tf[e] = 299792458.0f / wl;
    }
    for (int e = tid; e < 104 * 16; e += NTHREADS) Qp[e] = 0.0f;
    for (int n = tid; n < NSTATE; n += NTHREADS) {
        int   fidx = n >> 2;
        float wl   = (fidx < 4) ? x[b * 20 + fidx] : sigwl[fidx - 4];
        losses[n]  = (c2 + c1 * wl + c0 * wl * wl) * LN;
        float p    = (n < 16) ? fabsf(x[b * 20 + 4 + n]) : 1.0e-4f;
        P0[n] = p; acc[n] = p; argv[n] = p;
    }
    __syncthreads();

    // ---- per-batch Raman gain matrix g (constant over the ODE) --------
    for (int e = tid; e < 112 * 104; e += NTHREADS) {
        int   i  = e / 104, j = e - i * 104;
        float gv = 0.0f;
        if (i < F_TOT) {
            float fi   = tf[i], fj = tf[j];
            float diff = fj - fi;                       // total_f[j]-total_f[i]
            float idxf = fabsf(diff) / DFv;
            int   i0   = (int)floorf(idxf);
            if (i0 < 0)   i0 = 0;
            if (i0 > 799) i0 = 799;
            float frac = idxf - (float)i0;
            float gq   = resp[i0] * (1.0f - frac) + resp[i0 + 1] * frac;
            gq = (idxf <= 800.0f) ? gq : 0.0f;
            gq = (diff < 0.0f) ? -gq : gq;
            float ratio = fi / fj;
            gq *= (ratio > 1.0f) ? ratio : 1.0f;
            gv = gq;
        }
        gmat[e] = gv;
    }
    __syncthreads();

    // ---- hoist the invariant A operand (g rows) into registers --------
    // Each lane owns row 16w + (lane&15), K-phase 2*hi: 52 floats = 26 v2f.
    const int row   = 16 * w + l15;
    const int rbase = row * 104;
    v2f areg[26];
    #pragma unroll
    for (int k = 0; k < 26; ++k) {
        int koff = 4 * k + 2 * hi;
        areg[k].x = gmat[rbase + koff];
        areg[k].y = gmat[rbase + koff + 1];
    }

    // ---- RK4 integration: 99 steps × 4 stages -------------------------
    for (int step = 0; step < 99; ++step) {
        #pragma unroll
        for (int s = 0; s < 4; ++s) {
            const float wk = (s == 0 || s == 3) ? (dz / 6.0f) : (dz / 3.0f);
            const float sc = (s == 2) ? dz : 0.5f * dz;

            __syncthreads();            // argv stable
            // Q[j,mi] = oi @ P  (mode mixing), into B-operand layout
            for (int e = tid; e < NSTATE; e += NTHREADS) {
                int j = e >> 2, mi = e & 3;
                const float* av = &argv[j << 2];
                Qp[j * 16 + mi] = oi_s[mi * 4 + 0] * av[0] +
                                  oi_s[mi * 4 + 1] * av[1] +
                                  oi_s[mi * 4 + 2] * av[2] +
                                  oi_s[mi * 4 + 3] * av[3];
            }
            __syncthreads();

            // R = g @ Q via 26-chunk f32 WMMA accumulation (K = 104);
            // A comes from registers, only B streams from LDS.
            v8f c = {};
            #pragma unroll
            for (int k = 0; k < 26; ++k) {
                int koff = 4 * k + 2 * hi;
                v2f bv;
                bv.x = Qp[koff * 16 + l15];
                bv.y = Qp[(koff + 1) * 16 + l15];
                c = __builtin_amdgcn_wmma_f32_16x16x4_f32(
                        false, areg[k], false, bv, (short)0, c, false, false);
            }

            // fold loss + RK4 update into the D write-back (cols 0..3 only)
            if (l15 < 4) {
                #pragma unroll
                for (int v = 0; v < 8; ++v) {
                    int r = 16 * w + v + 8 * hi;
                    if (r < F_TOT) {
                        int   idx = (r << 2) + l15;
                        float av  = argv[idx];
                        float kx  = (c[v] - losses[idx]) * av;  // (-loss + G·P)·P
                        float an  = acc[idx] + wk * kx;
                        acc[idx]  = an;
                        if (s < 3) {
                            argv[idx] = P0[idx] + sc * kx;
                        } else {
                            P0[idx] = an;  argv[idx] = an;
                        }
                    }
                }
            }
        }
    }
    __syncthreads();

    // ---- emit signal powers: state elements 16..415 -------------------
    for (int n = tid; n < 400; n += NTHREADS) out[b * 400 + n] = P0[16 + n];
}

extern "C" void kernel_launch(void* const* d_in, const int* in_sizes, int n_in,
                              void* d_out, int out_size, void* d_ws, size_t ws_size,
                              hipStream_t stream) {
    const float* x     = (const float*)d_in[0];
    const float* rr    = (const float*)d_in[1];
    const float* oi    = (const float*)d_in[2];
    const float* sigwl = (const float*)d_in[3];
    const float* lc    = (const float*)d_in[4];
    float*       out   = (float*)d_out;
    (void)in_sizes; (void)n_in; (void)out_size; (void)d_ws; (void)ws_size;
    MMFRamanAmplifier_kernel<<<dim3(256), dim3(NTHREADS), 0, stream>>>(
        x, rr, oi, sigwl, lc, out);
}